// Attn_62380105008069
// MI455X (gfx1250) — compile-verified
//
#include <hip/hip_runtime.h>
#include <hip/hip_bf16.h>

typedef float v2f __attribute__((ext_vector_type(2)));
typedef float v8f __attribute__((ext_vector_type(8)));

#define BATCH 32
#define SEQ   4096
#define HID   512

// -----------------------------------------------------------------------------
// Kernel 1: v[b, n] = sum_o hidden[b, o] * W[o, n]   (i.e. v = hidden @ W)
// A = hidden [M=32, K=512], B = W [K=512, N=512], D = v [32, 512]
// One wave (32 threads) computes one 16x16 tile via V_WMMA_F32_16X16X4_F32.
// grid = (N/16=32, M/16=2), block = 32 (EXEC all ones, as WMMA requires).
// -----------------------------------------------------------------------------
__global__ void vmat_wmma(const float* __restrict__ hidden,
                          const float* __restrict__ W,
                          float* __restrict__ v) {
    const int lane = threadIdx.x;        // 0..31
    const int half = lane >> 4;          // 0: lanes 0-15, 1: lanes 16-31
    const int l    = lane & 15;
    const int n0   = blockIdx.x * 16;
    const int m0   = blockIdx.y * 16;

    v8f acc = {};
    for (int k0 = 0; k0 < HID; k0 += 4) {
        // A 16x4 f32 layout: VGPR0 = K {0 | 2}, VGPR1 = K {1 | 3} per lane half
        const int ka = k0 + 2 * half;
        v2f a, b;
        a.x = hidden[(m0 + l) * HID + ka + 0];
        a.y = hidden[(m0 + l) * HID + ka + 1];
        // B 4x16 f32: N striped across lanes, K across VGPR + lane half
        b.x = W[(ka + 0) * HID + n0 + l];
        b.y = W[(ka + 1) * HID + n0 + l];
        acc = __builtin_amdgcn_wmma_f32_16x16x4_f32(
            /*neg_a=*/false, a, /*neg_b=*/false, b,
            /*c_mod=*/(short)0, acc, /*reuse_a=*/false, /*reuse_b=*/false);
    }

    // D 16x16 f32 layout: VGPR j -> row M = j (lanes 0-15) / M = 8+j (lanes 16-31)
    const int mbase = m0 + (half ? 8 : 0);
    #pragma unroll
    for (int j = 0; j < 8; ++j) {
        v[(mbase + j) * HID + n0 + l] = acc[j];
    }
}

// -----------------------------------------------------------------------------
// Kernel 2: scores[b, s] = enc[b, s, :] . v[b, :]
// Memory-bound: streams 256 MB of enc exactly once with float4 loads.
// grid = (SEQ/128, BATCH), block = 256 (8 wave32s); one wave per row.
// -----------------------------------------------------------------------------
__global__ void scores_dot(const float* __restrict__ enc,
                           const float* __restrict__ v,
                           float* __restrict__ scores) {
    const int b      = blockIdx.y;
    const int s_base = blockIdx.x * 128;
    const int lane   = threadIdx.x & 31;
    const int wave   = threadIdx.x >> 5;

    // Each lane caches its 16 v-values (4 x float4) — same for all waves,
    // served from L2/WGP$ once per block.
    float4 vv[4];
    #pragma unroll
    for (int k = 0; k < 4; ++k) {
        vv[k] = *(const float4*)(v + b * HID + 4 * lane + 128 * k);
    }

    const size_t enc_b = (size_t)b * SEQ * HID;

    for (int r = wave; r < 128; r += 8) {
        const int s = s_base + r;
        const float4* row = (const float4*)(enc + enc_b + (size_t)s * HID);
        // hint the next row for this wave into the cache hierarchy
        if (r + 8 < 128) {
            __builtin_prefetch(enc + enc_b + (size_t)(s + 8) * HID + 4 * lane, 0, 1);
        }
        float acc = 0.0f;
        #pragma unroll
        for (int k = 0; k < 4; ++k) {
            const float4 e = row[lane + 32 * k];
            acc += e.x * vv[k].x + e.y * vv[k].y + e.z * vv[k].z + e.w * vv[k].w;
        }
        // wave32 all-reduce
        #pragma unroll
        for (int off = 16; off >= 1; off >>= 1) {
            acc += __shfl_xor(acc, off, 32);
        }
        if (lane == 0) {
            scores[b * SEQ + s] = acc;
        }
    }
}

// -----------------------------------------------------------------------------
// Kernel 3: out[b, 0, s] = softmax_s(scores[b, :])   (numerically stable)
// grid = BATCH, block = 256; each thread strides 16 elements of S=4096.
// -----------------------------------------------------------------------------
__global__ void softmax_rows(const float* __restrict__ scores,
                             float* __restrict__ out) {
    __shared__ float red[256];
    const int b   = blockIdx.x;
    const int tid = threadIdx.x;
    const float* row = scores + b * SEQ;

    // 1) max
    float m = -3.402823466e+38f;
    for (int i = tid; i < SEQ; i += 256) {
        m = fmaxf(m, row[i]);
    }
    red[tid] = m;
    __syncthreads();
    for (int w = 128; w >= 1; w >>= 1) {
        if (tid < w) red[tid] = fmaxf(red[tid], red[tid + w]);
        __syncthreads();
    }
    const float rmax = red[0];
    __syncthreads();

    // 2) sum of exp
    float ssum = 0.0f;
    for (int i = tid; i < SEQ; i += 256) {
        ssum += expf(row[i] - rmax);
    }
    red[tid] = ssum;
    __syncthreads();
    for (int w = 128; w >= 1; w >>= 1) {
        if (tid < w) red[tid] += red[tid + w];
        __syncthreads();
    }
    const float inv = 1.0f / red[0];
    __syncthreads();

    // 3) normalize
    for (int i = tid; i < SEQ; i += 256) {
        out[b * SEQ + i] = expf(row[i] - rmax) * inv;
    }
}

// -----------------------------------------------------------------------------
// Launcher. Inputs (setup_inputs order): hidden [1,B,H], encoder_outputs
// [B,S,H], W_attn [H,H], b_attn [H]. Output: attn [B,1,S] flat = B*S floats.
// Note: b_attn contributes a per-batch constant to scores, which softmax's
// shift invariance cancels exactly, so it is not needed.
// -----------------------------------------------------------------------------
extern "C" void kernel_launch(void* const* d_in, const int* in_sizes, int n_in,
                              void* d_out, int out_size, void* d_ws, size_t ws_size,
                              hipStream_t stream) {
    const float* hidden = (const float*)d_in[0];
    const float* enc    = (const float*)d_in[1];
    const float* W      = (const float*)d_in[2];
    // d_in[3] = b_attn: unused (softmax shift-invariant)

    float* out    = (float*)d_out;
    float* v      = (float*)d_ws;            // BATCH*HID floats   (64 KB)
    float* scores = v + BATCH * HID;         // BATCH*SEQ floats  (512 KB)

    vmat_wmma<<<dim3(HID / 16, BATCH / 16), 32, 0, stream>>>(hidden, W, v);
    scores_dot<<<dim3(SEQ / 128, BATCH), 256, 0, stream>>>(enc, v, scores);
    softmax_rows<<<BATCH, 256, 0, stream>>>(scores, out);
}